// Memory_63599875719529
// MI455X (gfx1250) — compile-verified
//
#include <hip/hip_runtime.h>

// Problem constants (from reference)
#define FEAT    512
#define MEMN    256
#define BATCHN  512
#define PQ      131072      // 64*2048
#define TOPK    16
#define EPSN    1e-8f

typedef __attribute__((ext_vector_type(2))) float v2f;
typedef __attribute__((ext_vector_type(8))) float v8f;

// ---------------------------------------------------------------------------
// Kernel 1: inverse key norms (256 rows x 512) -> invk[256]
// ---------------------------------------------------------------------------
__global__ void key_norms_kernel(const float* __restrict__ keys,
                                 float* __restrict__ invk) {
    int m = blockIdx.x * blockDim.x + threadIdx.x;
    if (m >= MEMN) return;
    const float* kr = keys + (size_t)m * FEAT;
    float s = 0.f;
    for (int i = 0; i < FEAT; ++i) { float v = kr[i]; s = fmaf(v, v, s); }
    invk[m] = 1.0f / fmaxf(sqrtf(s), EPSN);
}

// ---------------------------------------------------------------------------
// Kernel 2: per-sample cosine sim (256 keys), top-16, clamp>=0, renormalize,
// scatter into dense W[b][256] in workspace. One 256-thread block per sample.
// ---------------------------------------------------------------------------
__global__ void __launch_bounds__(256)
topk_weights_kernel(const float* __restrict__ features,
                    const float* __restrict__ keys,
                    const float* __restrict__ invk,
                    float* __restrict__ W) {
    __shared__ float sim[MEMN];
    __shared__ float rv[MEMN];
    __shared__ int   ri[MEMN];
    __shared__ float topv[TOPK];
    __shared__ int   topi[TOPK];
    __shared__ float wsum;

    const int b = blockIdx.x;
    const int t = threadIdx.x;               // t == key index m
    const float* fr = features + (size_t)b * FEAT;
    const float* kr = keys + (size_t)t * FEAT;

    float dot = 0.f, f2 = 0.f;
    for (int i = 0; i < FEAT; ++i) {
        float fv = fr[i];
        dot = fmaf(fv, kr[i], dot);
        f2  = fmaf(fv, fv, f2);
    }
    float invf = 1.0f / fmaxf(sqrtf(f2), EPSN);
    sim[t] = dot * invf * invk[t];
    __syncthreads();

    // 16 sequential block-wide argmax reductions
    for (int round = 0; round < TOPK; ++round) {
        rv[t] = sim[t]; ri[t] = t;
        __syncthreads();
        for (int s = MEMN / 2; s > 0; s >>= 1) {
            if (t < s) {
                if (rv[t + s] > rv[t]) { rv[t] = rv[t + s]; ri[t] = ri[t + s]; }
            }
            __syncthreads();
        }
        if (t == 0) {
            topv[round] = rv[0];
            topi[round] = ri[0];
            sim[ri[0]]  = -3.0e38f;          // remove winner
        }
        __syncthreads();
    }

    if (t == 0) {
        float s = 0.f;
        for (int i = 0; i < TOPK; ++i) {
            float v = fmaxf(topv[i], 0.f);
            topv[i] = v;
            s += v;
        }
        wsum = s;
    }
    __syncthreads();

    float* Wrow = W + (size_t)b * MEMN;
    Wrow[t] = 0.f;                           // zero dense row
    __syncthreads();
    if (t < TOPK) {
        Wrow[topi[t]] = topv[t] / wsum;      // distinct indices by construction
    }
}

// ---------------------------------------------------------------------------
// Kernel 3: ctx (512 x 131072) = W (512 x 256) @ memory (256 x 131072), f32,
// via V_WMMA_F32_16X16X4_F32. Block = 8 wave32 = one 16-row M-tile x 128 cols.
// A tile staged in LDS (pitch 260 -> conflict-free ds_load_b64 fragments).
// B streamed from global (memory bank stays L2-resident: 128MB < 192MB L2).
// C written with non-temporal stores (don't pollute L2 with the 256MB output).
// ---------------------------------------------------------------------------
__global__ void __launch_bounds__(256)
ctx_wmma_kernel(const float* __restrict__ W,
                const float* __restrict__ mem,
                float* __restrict__ out) {
    __shared__ float Alds[16 * 260];

    const int m0   = blockIdx.x * 16;        // 32 M-tiles (x-fastest => L2 reuse of B)
    const int n0g  = blockIdx.y * 128;       // 1024 column groups
    const int wave = threadIdx.x >> 5;
    const int lane = threadIdx.x & 31;
    const int n0   = n0g + wave * 16;        // this wave's 16 output columns

    // Stage A = W[m0:m0+16, 0:256] into LDS
    for (int i = threadIdx.x; i < 16 * MEMN; i += 256) {
        int r = i >> 8, k = i & (MEMN - 1);
        Alds[r * 260 + k] = W[(size_t)(m0 + r) * MEMN + k];
    }
    __syncthreads();

    // wave32 WMMA fragment mapping (16x16x4 f32):
    //   lanes 0-15  hold K = k+0 (v0) and k+1 (v1)
    //   lanes 16-31 hold K = k+2 (v0) and k+3 (v1)
    const int half = lane >> 4;              // 0 | 1
    const int l15  = lane & 15;              // row (A) / col (B,C)
    const int koff = half << 1;              // 0 | 2
    const float* Arow = &Alds[l15 * 260];
    const float* Bcol = mem + n0 + l15;

    v8f acc = {};
    #pragma unroll 4
    for (int k = 0; k < MEMN; k += 4) {
        v2f a;
        a.x = Arow[k + koff];
        a.y = Arow[k + koff + 1];
        v2f bv;
        bv.x = Bcol[(size_t)(k + koff)     * PQ];
        bv.y = Bcol[(size_t)(k + koff + 1) * PQ];
        // D = A(16x4) * B(4x16) + C, f32 accumulate
        acc = __builtin_amdgcn_wmma_f32_16x16x4_f32(
                  /*neg_a=*/false, a, /*neg_b=*/false, bv,
                  /*c_mod=*/(short)0, acc, /*reuse_a=*/false, /*reuse_b=*/false);
    }

    // C/D layout: VGPR v -> rows v (lanes 0-15) and v+8 (lanes 16-31), col = l15
    float* orow = out + ((size_t)(m0 + half * 8)) * PQ + n0 + l15;
    #pragma unroll
    for (int v = 0; v < 8; ++v) {
        __builtin_nontemporal_store(acc[v], &orow[(size_t)v * PQ]);
    }
}

// ---------------------------------------------------------------------------
extern "C" void kernel_launch(void* const* d_in, const int* in_sizes, int n_in,
                              void* d_out, int out_size, void* d_ws, size_t ws_size,
                              hipStream_t stream) {
    const float* features = (const float*)d_in[0];   // (512, 512)
    const float* keys     = (const float*)d_in[1];   // (256, 512)
    const float* mem      = (const float*)d_in[2];   // (256, 64, 2048)
    float* out = (float*)d_out;                      // (512, 64, 2048)

    float* W    = (float*)d_ws;                      // 512*256 f32
    float* invk = W + (size_t)BATCHN * MEMN;         // 256 f32

    key_norms_kernel<<<1, 256, 0, stream>>>(keys, invk);
    topk_weights_kernel<<<BATCHN, 256, 0, stream>>>(features, keys, invk, W);

    dim3 grid(BATCHN / 16, PQ / 128);                // (32, 1024)
    ctx_wmma_kernel<<<grid, 256, 0, stream>>>(W, mem, out);
}